// SyntaxGAT_19653770347184
// MI455X (gfx1250) — compile-verified
//
#include <hip/hip_runtime.h>
#include <hip/hip_bf16.h>
#include <math.h>

typedef float v2f __attribute__((ext_vector_type(2)));
typedef float v8f __attribute__((ext_vector_type(8)));

#define S_ 96
#define DIN 768

__device__ __forceinline__ float sigmoidf_(float x) { return 1.0f / (1.0f + expf(-x)); }

// ---------------------------------------------------------------------------
// dep_weights[b,i,j] = sigmoid(dep_emb[type]·att_w + att_b) * dep_matrix
// ---------------------------------------------------------------------------
__global__ void depw_kernel(const int* __restrict__ types, const float* __restrict__ dmat,
                            const float* __restrict__ emb, const float* __restrict__ aw,
                            const float* __restrict__ ab, float* __restrict__ depw, int n) {
    int e = blockIdx.x * blockDim.x + threadIdx.x;
    if (e >= n) return;
    int t = types[e];
    const float* ep = emb + t * 6;
    float acc = ab[0];
#pragma unroll
    for (int k = 0; k < 6; ++k) acc += ep[k] * aw[k];
    depw[e] = sigmoidf_(acc) * dmat[e];
}

// ---------------------------------------------------------------------------
// h_in[b,i,d] = sum_j h[b,i,j,d]*w[b,i,j]   (blockIdx.y == 0)
// h_out[b,j,d] = sum_i h[b,i,j,d]*w[b,i,j]  (blockIdx.y == 1)
// ---------------------------------------------------------------------------
__global__ void synred_kernel(const float* __restrict__ h, const float* __restrict__ depw,
                              float* __restrict__ hin, float* __restrict__ hout) {
    int bs = blockIdx.x;          // b*S + (i or j)
    int b = bs / S_;
    int x = bs % S_;
    if (blockIdx.y == 0) {
        const float* hp = h + ((size_t)(b * S_ + x) * S_) * DIN;
        const float* wp = depw + (size_t)(b * S_ + x) * S_;
        for (int d = threadIdx.x; d < DIN; d += blockDim.x) {
            float acc = 0.f;
            for (int j = 0; j < S_; ++j) acc += hp[(size_t)j * DIN + d] * wp[j];
            hin[(size_t)bs * DIN + d] = acc;
        }
    } else {
        const float* hp = h + ((size_t)b * S_ * S_ + x) * DIN;
        const float* wp = depw + (size_t)b * S_ * S_ + x;
        for (int d = threadIdx.x; d < DIN; d += blockDim.x) {
            float acc = 0.f;
            for (int i = 0; i < S_; ++i) acc += hp[(size_t)i * S_ * DIN + d] * wp[(size_t)i * S_];
            hout[(size_t)bs * DIN + d] = acc;
        }
    }
}

// ---------------------------------------------------------------------------
// X[r, 0:768]    = sem (5-point weighted stencil over (i,j))
// X[r, 768:1536] = syn = h_in[b,i,:] + h_out[b,j,:]
// ---------------------------------------------------------------------------
__device__ __forceinline__ void fma4(float4& a, float w, const float4 v) {
    a.x += w * v.x; a.y += w * v.y; a.z += w * v.z; a.w += w * v.w;
}

__global__ void buildx_kernel(const float* __restrict__ h, const float* __restrict__ a_simi,
                              const float* __restrict__ t_simi, const float* __restrict__ hin,
                              const float* __restrict__ hout, float* __restrict__ X) {
    int r = blockIdx.x;
    int b = r / (S_ * S_);
    int ij = r % (S_ * S_);
    int i = ij / S_, j = ij % S_;
    int q = threadIdx.x;   // 0..191 (float4 index into 768)

    float4 sem = ((const float4*)(h + (size_t)r * DIN))[q];
    if (j > 0)      fma4(sem, t_simi[b * S_ + j - 1], ((const float4*)(h + (size_t)(r - 1) * DIN))[q]);
    if (j < S_ - 1) fma4(sem, t_simi[b * S_ + j + 1], ((const float4*)(h + (size_t)(r + 1) * DIN))[q]);
    if (i > 0)      fma4(sem, a_simi[b * S_ + i - 1], ((const float4*)(h + (size_t)(r - S_) * DIN))[q]);
    if (i < S_ - 1) fma4(sem, a_simi[b * S_ + i + 1], ((const float4*)(h + (size_t)(r + S_) * DIN))[q]);
    ((float4*)(X + (size_t)r * 2 * DIN))[q] = sem;

    float4 si = ((const float4*)(hin + (size_t)(b * S_ + i) * DIN))[q];
    float4 so = ((const float4*)(hout + (size_t)(b * S_ + j) * DIN))[q];
    float4 syn = make_float4(si.x + so.x, si.y + so.y, si.z + so.z, si.w + so.w);
    ((float4*)(X + (size_t)r * 2 * DIN + DIN))[q] = syn;
}

// ---------------------------------------------------------------------------
// Tiled fp32 WMMA GEMM: out(M,N) = A(M,K) @ Bm(K,N)  [+ epilogue]
// Block tile 128x64, BK=16, 256 threads = 8 wave32 in a 4x2 wave grid;
// each wave owns a 32x32 tile -> 4 v_wmma_f32_16x16x4_f32 accumulators.
// A frags: contiguous (k,k+1) pair in row-major sA  -> single ds b64.
// B frags: k-pair interleaved layout sB[k/2][col][k&1] -> single ds b64.
// EPI==0: gated fusion epilogue  u = g*sem + (1-g)*syn + resid   (N==768)
// EPI==1: out = acc + bias
// ---------------------------------------------------------------------------
template <int EPI>
__global__ __launch_bounds__(256) void gemm_wmma_kernel(
    const float* __restrict__ A, const float* __restrict__ Bm,
    float* __restrict__ out, int K, int N,
    const float* __restrict__ bias,
    const float* __restrict__ X,       // EPI==0: concat [sem|syn], stride 1536
    const float* __restrict__ resid) { // EPI==0: residual h, stride 768
    __shared__ __align__(16) float sA[128][20];     // [row][k], padded, 16B aligned rows
    __shared__ __align__(16) float sB[8][66][2];    // [k/2][col][k&1], pairs contiguous

    const int tid = threadIdx.x;
    const int lane = tid & 31, wid = tid >> 5;
    const int waveM = wid & 3, waveN = wid >> 2;   // waves: 4 row groups x 2 col groups
    const int lm = lane & 15, lh = lane >> 4;

    const int rowBase = blockIdx.y * 128;
    const int colBase = blockIdx.x * 64;

    v8f acc00 = {}, acc01 = {}, acc10 = {}, acc11 = {};

    // cooperative loads: A 128x16 (2 float4 per thread), B 16x64 (1 float4)
    const int aRow = tid >> 2, aK = (tid & 3) * 4;
    const int bK = tid >> 4, bCol = (tid & 15) * 4;

    for (int k0 = 0; k0 < K; k0 += 16) {
        float4 av0 = *(const float4*)(A + (size_t)(rowBase + aRow) * K + k0 + aK);
        float4 av1 = *(const float4*)(A + (size_t)(rowBase + aRow + 64) * K + k0 + aK);
        float4 bv  = *(const float4*)(Bm + (size_t)(k0 + bK) * N + colBase + bCol);
        __syncthreads();
        *(float4*)&sA[aRow][aK]      = av0;
        *(float4*)&sA[aRow + 64][aK] = av1;
        sB[bK >> 1][bCol + 0][bK & 1] = bv.x;
        sB[bK >> 1][bCol + 1][bK & 1] = bv.y;
        sB[bK >> 1][bCol + 2][bK & 1] = bv.z;
        sB[bK >> 1][bCol + 3][bK & 1] = bv.w;
        __syncthreads();
#pragma unroll
        for (int kk = 0; kk < 16; kk += 4) {
            const int ks = kk + 2 * lh;            // lanes 16-31 take K+2,K+3
            v2f a0 = *(const v2f*)&sA[waveM * 32 + lm][ks];
            v2f a1 = *(const v2f*)&sA[waveM * 32 + 16 + lm][ks];
            v2f b0 = *(const v2f*)&sB[ks >> 1][waveN * 32 + lm][0];
            v2f b1 = *(const v2f*)&sB[ks >> 1][waveN * 32 + 16 + lm][0];
            acc00 = __builtin_amdgcn_wmma_f32_16x16x4_f32(false, a0, false, b0,
                                                          (short)0, acc00, false, false);
            acc01 = __builtin_amdgcn_wmma_f32_16x16x4_f32(false, a0, false, b1,
                                                          (short)0, acc01, false, false);
            acc10 = __builtin_amdgcn_wmma_f32_16x16x4_f32(false, a1, false, b0,
                                                          (short)0, acc10, false, false);
            acc11 = __builtin_amdgcn_wmma_f32_16x16x4_f32(false, a1, false, b1,
                                                          (short)0, acc11, false, false);
        }
    }

    // C/D layout: VGPR v -> row = base + v + 8*(lane/16), col = base + (lane%16)
#pragma unroll
    for (int rt = 0; rt < 2; ++rt) {
#pragma unroll
        for (int ct = 0; ct < 2; ++ct) {
            const v8f& acc = (rt == 0) ? ((ct == 0) ? acc00 : acc01)
                                       : ((ct == 0) ? acc10 : acc11);
            int rG0 = rowBase + waveM * 32 + rt * 16 + 8 * lh;
            int c   = colBase + waveN * 32 + ct * 16 + lm;
#pragma unroll
            for (int v = 0; v < 8; ++v) {
                int r = rG0 + v;
                float val = acc[v];
                if (EPI == 0) {
                    float g = sigmoidf_(val + bias[c]);
                    float sem = X[(size_t)r * 1536 + c];
                    float syn = X[(size_t)r * 1536 + 768 + c];
                    out[(size_t)r * 768 + c] =
                        g * sem + (1.0f - g) * syn + resid[(size_t)r * 768 + c];
                } else {
                    out[(size_t)r * N + c] = val + bias[c];
                }
            }
        }
    }
}

// ---------------------------------------------------------------------------
// T5 layernorm (+1e-12) then ReLU, one block per row
// ---------------------------------------------------------------------------
__global__ void lnrelu_kernel(const float* __restrict__ y, const float* __restrict__ w,
                              float* __restrict__ out, int N) {
    __shared__ float red[256];
    int r = blockIdx.x;
    const float* yp = y + (size_t)r * N;
    float ss = 0.f;
    for (int c = threadIdx.x; c < N; c += 256) { float v = yp[c]; ss += v * v; }
    red[threadIdx.x] = ss;
    __syncthreads();
    for (int s = 128; s > 0; s >>= 1) {
        if (threadIdx.x < s) red[threadIdx.x] += red[threadIdx.x + s];
        __syncthreads();
    }
    float inv = rsqrtf(red[0] / (float)N + 1e-12f);
    for (int c = threadIdx.x; c < N; c += 256) {
        float v = yp[c] * inv * w[c];
        out[(size_t)r * N + c] = v > 0.f ? v : 0.f;
    }
}

// ---------------------------------------------------------------------------
extern "C" void kernel_launch(void* const* d_in, const int* in_sizes, int n_in,
                              void* d_out, int out_size, void* d_ws, size_t ws_size,
                              hipStream_t stream) {
    const float* table   = (const float*)d_in[0];
    const float* a_simi  = (const float*)d_in[1];
    const float* t_simi  = (const float*)d_in[2];
    const float* dep_mat = (const float*)d_in[3];
    const int*   dep_typ = (const int*)d_in[4];
    const float* d1w = (const float*)d_in[5];
    const float* d1b = (const float*)d_in[6];
    const float* d2w = (const float*)d_in[7];
    const float* d2b = (const float*)d_in[8];
    const float* demb = (const float*)d_in[9];
    const float* daw  = (const float*)d_in[10];
    const float* dab  = (const float*)d_in[11];
    const float* gw   = (const float*)d_in[12];
    const float* gb   = (const float*)d_in[13];
    const float* n1w  = (const float*)d_in[14];
    const float* n2w  = (const float*)d_in[15];

    const int B = 4, S = 96, D = 768;
    const int M = B * S * S;                        // 36864 (divisible by 128)

    float* ws = (float*)d_ws;
    float* depw = ws;  ws += M;
    float* hin  = ws;  ws += (size_t)B * S * D;
    float* hout = ws;  ws += (size_t)B * S * D;
    float* X    = ws;  ws += (size_t)M * 2 * D;     // [sem | syn]
    float* u    = ws;  ws += (size_t)M * D;         // fused + residual
    float* ybuf = ws;  ws += (size_t)M * D;         // pre-layernorm GEMM output
    float* h1   = ws;  ws += (size_t)M * D;         // layer-1 output

    depw_kernel<<<(M + 255) / 256, 256, 0, stream>>>(dep_typ, dep_mat, demb, daw, dab,
                                                     depw, M);

    for (int l = 0; l < 2; ++l) {
        const float* h  = (l == 0) ? table : h1;
        const float* dw = (l == 0) ? d1w : d2w;
        const float* db = (l == 0) ? d1b : d2b;
        const float* nw = (l == 0) ? n1w : n2w;
        float* lnout    = (l == 0) ? h1 : (float*)d_out;
        const int Nout  = (l == 0) ? 768 : 256;

        synred_kernel<<<dim3(B * S, 2), 256, 0, stream>>>(h, depw, hin, hout);
        buildx_kernel<<<M, 192, 0, stream>>>(h, a_simi, t_simi, hin, hout, X);
        // gate GEMM + fusion + residual:  u = fuse(X @ gate_w)
        gemm_wmma_kernel<0><<<dim3(768 / 64, M / 128), 256, 0, stream>>>(
            X, gw, u, 1536, 768, gb, X, h);
        // dense GEMM:  ybuf = u @ dense_w + b
        gemm_wmma_kernel<1><<<dim3(Nout / 64, M / 128), 256, 0, stream>>>(
            u, dw, ybuf, 768, Nout, db, nullptr, nullptr);
        lnrelu_kernel<<<M, 256, 0, stream>>>(ybuf, nw, lnout, Nout);
    }
}